// RSSM_51196010168456
// MI455X (gfx1250) — compile-verified
//
#include <hip/hip_runtime.h>
#include <hip/hip_bf16.h>
#include <math.h>

// ---------------------------------------------------------------------------
// RSSM scan for MI455X (gfx1250, wave32, WMMA).
// Strategy:
//  * Hoist embed @ Wr1[200:,:] out of the scan -> one big f32 WMMA GEMM
//    (memory bound at 23.3 TB/s; f32 WMMA keeps full precision for free).
//  * Persistent single-workgroup (32 waves) sequential scan with all
//    activations in LDS (~250KB of 320KB) and weights pre-packed into
//    WMMA B-fragment order (coalesced float2-per-lane loads, L2 resident).
// ---------------------------------------------------------------------------

typedef float v2f __attribute__((ext_vector_type(2)));
typedef float v8f __attribute__((ext_vector_type(8)));

#define B_     64
#define T_     256
#define STOCH_ 30
#define DETER_ 200
#define HID_   200
#define ADIM_  12
#define EDIM_  1024
#define OUTC_  580

// ---- workspace layout (float element offsets) ----
#define EPROJ_OFF  0
#define EPROJ_SZ   (B_ * T_ * HID_)        // 3,276,800
#define WE_P_OFF   (EPROJ_OFF + EPROJ_SZ)
#define WE_P_SZ    (13 * 11 * 64)          // K=42->44, N=200->208
#define WI_P_OFF   (WE_P_OFF + WE_P_SZ)
#define WI_P_SZ    (38 * 50 * 64)          // K=200, N=600->608
#define WH_P_OFF   (WI_P_OFF + WI_P_SZ)
#define WH_P_SZ    (38 * 50 * 64)
#define WT1_P_OFF  (WH_P_OFF + WH_P_SZ)
#define WT1_P_SZ   (13 * 50 * 64)          // K=200, N=200->208
#define WT2_P_OFF  (WT1_P_OFF + WT1_P_SZ)
#define WT2_P_SZ   (4 * 50 * 64)           // K=200, N=60->64
#define WR1H_P_OFF (WT2_P_OFF + WT2_P_SZ)
#define WR1H_P_SZ  (13 * 50 * 64)          // Wr1 rows 0..199
#define WR1E_P_OFF (WR1H_P_OFF + WR1H_P_SZ)
#define WR1E_P_SZ  (13 * 256 * 64)         // Wr1 rows 200..1223 (K=1024)
#define WR2_P_OFF  (WR1E_P_OFF + WR1E_P_SZ)
#define WR2_P_SZ   (4 * 50 * 64)

// ---- LDS layout for sequential kernel (float offsets) ----
#define S_STOCH_OFF 0                        // [64][48]: stoch|action|zero-pad
#define S_H_OFF     (64 * 48)                // [64][200]
#define S_X_OFF     (S_H_OFF + 64 * 200)     // [64][200]
#define S_Z_OFF     (S_X_OFF + 64 * 200)     // [64][200]
#define S_RH_OFF    (S_Z_OFF + 64 * 200)     // [64][200]
#define S_Y_OFF     (S_RH_OFF + 64 * 200)    // [64][64]
#define S_YQ_OFF    (S_Y_OFF + 64 * 64)      // [64][64]
#define SMEM_FLOATS (S_YQ_OFF + 64 * 64)     // 62,464 floats = 249,856 B

// ---------------------------------------------------------------------------
// math helpers
// ---------------------------------------------------------------------------
__device__ __forceinline__ float elu1(float v) { return v > 0.f ? v : expf(v) - 1.f; }
__device__ __forceinline__ float sgm(float v)  { return 1.f / (1.f + expf(-v)); }
__device__ __forceinline__ float sp1(float v)  { return v > 20.f ? v : log1pf(expf(v)); }

// ---------------------------------------------------------------------------
// One 16x16 tile K-loop: acc += A[m0:m0+16, 0:4*Kg] * Bp(tile nt)
// A row-major (LDS or global), 8B-aligned float2 per lane.
// Bp packed: element ((nt*Kg + kg)*32 + lane)*2 + j  holds
//            W[4*kg + 2*(lane/16) + j][nt*16 + lane%16]
// ---------------------------------------------------------------------------
__device__ __forceinline__ v8f frag_mm(const float* __restrict__ A, int lda, int m0,
                                       const float* __restrict__ Bp, int Kg, int nt,
                                       v8f acc, int lm, int half, int lane) {
  const float* ap = A + (size_t)(m0 + lm) * lda + 2 * half;
  const float* bp = Bp + ((size_t)nt * Kg) * 64 + lane * 2;
  for (int kg = 0; kg < Kg; ++kg) {
    v2f a = *(const v2f*)(ap + kg * 4);
    v2f b = *(const v2f*)(bp + kg * 64);
    acc = __builtin_amdgcn_wmma_f32_16x16x4_f32(false, a, false, b, (short)0, acc,
                                                false, false);
  }
  return acc;
}

// ---------------------------------------------------------------------------
// Repack W[K x N] (row-major, leading dim ldw, starting at row rowOff) into
// WMMA B-fragment order. Out-of-range -> 0 (handles K/N padding).
// ---------------------------------------------------------------------------
__global__ void repack_w(const float* __restrict__ W, float* __restrict__ Wp,
                         int K, int rowOff, int ldw, int N, int Kg, int Nt) {
  int total = Nt * Kg * 64;
  for (int idx = blockIdx.x * blockDim.x + threadIdx.x; idx < total;
       idx += gridDim.x * blockDim.x) {
    int j    = idx & 1;
    int lane = (idx >> 1) & 31;
    int g    = (idx >> 6) % Kg;
    int nt   = idx / (Kg * 64);
    int sk = 4 * g + 2 * (lane >> 4) + j;
    int sn = nt * 16 + (lane & 15);
    float v = 0.f;
    if (sk < K && sn < N) v = W[(size_t)(sk + rowOff) * ldw + sn];
    Wp[idx] = v;
  }
}

// ---------------------------------------------------------------------------
// eproj[r, n] = embed[r, :] @ Wr1[200:1224, n],  r = b*T + t  (16384 rows)
// One wave per 16x16 output tile; K=1024 -> 256 f32 WMMAs per tile.
// ---------------------------------------------------------------------------
__global__ void __launch_bounds__(256)
eproj_kernel(const float* __restrict__ embed, const float* __restrict__ Wp,
             float* __restrict__ eproj) {
  const int wave = threadIdx.x >> 5, lane = threadIdx.x & 31;
  const int lm = lane & 15, half = lane >> 4;
  const int NT = 13, KG = 256;
  int tile = blockIdx.x * (blockDim.x >> 5) + wave;
  if (tile >= 1024 * NT) return;                       // wave-uniform
  int mt = tile / NT, nt = tile % NT;
  int m0 = mt << 4;

  const float* ap = embed + (size_t)(m0 + lm) * EDIM_ + 2 * half;
  const float* bp = Wp + ((size_t)nt * KG) * 64 + lane * 2;
  v8f acc = {};
  for (int kg = 0; kg < KG; ++kg) {
    v2f a = *(const v2f*)(ap + kg * 4);
    v2f b = *(const v2f*)(bp + kg * 64);
    acc = __builtin_amdgcn_wmma_f32_16x16x4_f32(false, a, false, b, (short)0, acc,
                                                false, false);
  }
  int n = (nt << 4) + lm;
  if (n < HID_) {
    for (int v = 0; v < 8; ++v) {
      int m = m0 + v + 8 * half;
      eproj[(size_t)m * HID_ + n] = acc[v];
    }
  }
}

// ---------------------------------------------------------------------------
// Persistent sequential scan: 1 workgroup, 32 waves, 250KB dynamic LDS.
// ---------------------------------------------------------------------------
__global__ void __launch_bounds__(1024)
rssm_seq(const float* __restrict__ action,
         const float* __restrict__ nprior, const float* __restrict__ npost,
         const float* __restrict__ be, const float* __restrict__ bg,
         const float* __restrict__ bt1, const float* __restrict__ bt2,
         const float* __restrict__ br1, const float* __restrict__ br2,
         const float* __restrict__ we_p, const float* __restrict__ wi_p,
         const float* __restrict__ wh_p, const float* __restrict__ wt1_p,
         const float* __restrict__ wt2_p, const float* __restrict__ wr1h_p,
         const float* __restrict__ wr2_p, const float* __restrict__ eproj,
         float* __restrict__ out) {
  extern __shared__ float sm[];
  float* s_stoch = sm + S_STOCH_OFF;   // [64][48]: 0..29 stoch, 30..41 act, pad 0
  float* s_h     = sm + S_H_OFF;       // [64][200]
  float* s_x     = sm + S_X_OFF;       // [64][200]  (x, later t1)
  float* s_z     = sm + S_Z_OFF;       // [64][200]
  float* s_rh    = sm + S_RH_OFF;      // [64][200]  (r*h, later r1)
  float* s_y     = sm + S_Y_OFF;       // [64][64]   prior head
  float* s_yq    = sm + S_YQ_OFF;      // [64][64]   posterior head

  const int tid  = threadIdx.x;
  const int wave = tid >> 5, lane = tid & 31;
  const int lm = lane & 15, half = lane >> 4;

  // init: stoch = 0 (incl. padding cols), h = 0
  for (int i = tid; i < 64 * 48; i += 1024) s_stoch[i] = 0.f;
  for (int i = tid; i < 64 * 200; i += 1024) s_h[i] = 0.f;
  __syncthreads();

#pragma unroll 1
  for (int t = 0; t < T_; ++t) {
    // stage action[b,t,:] into s_stoch cols 30..41
    if (tid < B_ * ADIM_) {
      int b = tid / ADIM_, j = tid % ADIM_;
      s_stoch[b * 48 + STOCH_ + j] = action[((size_t)(b * T_ + t)) * ADIM_ + j];
    }
    __syncthreads();

    // Phase 1: x = elu([stoch|act] @ We + be)      [64 x 44] x [44 x 200]
    for (int tile = wave; tile < 52; tile += 32) {
      int m0 = (tile & 3) << 4, nt = tile >> 2;
      int n = (nt << 4) + lm;
      float bias = (n < HID_) ? be[n] : 0.f;
      v8f acc; for (int v = 0; v < 8; ++v) acc[v] = bias;
      acc = frag_mm(s_stoch, 48, m0, we_p, 11, nt, acc, lm, half, lane);
      if (n < HID_)
        for (int v = 0; v < 8; ++v)
          s_x[(m0 + v + 8 * half) * 200 + n] = elu1(acc[v]);
    }
    __syncthreads();

    // Phase 2: zr = sigmoid(x@wi[:,:400] + h@wh[:,:400] + bg[:400])
    //          z -> s_z ; r -> s_rh = r*h
    for (int tile = wave; tile < 100; tile += 32) {
      int m0 = (tile & 3) << 4, nt = tile >> 2;   // nt 0..24, n < 400
      int n = (nt << 4) + lm;
      float bias = bg[n];
      v8f acc; for (int v = 0; v < 8; ++v) acc[v] = bias;
      acc = frag_mm(s_x, 200, m0, wi_p, 50, nt, acc, lm, half, lane);
      acc = frag_mm(s_h, 200, m0, wh_p, 50, nt, acc, lm, half, lane);
      for (int v = 0; v < 8; ++v) {
        int m = m0 + v + 8 * half;
        float s = sgm(acc[v]);
        if (n < 200) s_z[m * 200 + n] = s;
        else         s_rh[m * 200 + (n - 200)] = s * s_h[m * 200 + (n - 200)];
      }
    }
    __syncthreads();

    // Phase 3: a = tanh(x@wi[:,400:] + (r*h)@wh[:,400:] + bg[400:])
    //          h = (1-z)*h + z*a   (in place)
    for (int tile = wave; tile < 52; tile += 32) {
      int m0 = (tile & 3) << 4, ntl = tile >> 2;
      int nt = 25 + ntl;
      int n = (ntl << 4) + lm;
      float bias = (n < 200) ? bg[400 + n] : 0.f;
      v8f acc; for (int v = 0; v < 8; ++v) acc[v] = bias;
      acc = frag_mm(s_x, 200, m0, wi_p, 50, nt, acc, lm, half, lane);
      acc = frag_mm(s_rh, 200, m0, wh_p, 50, nt, acc, lm, half, lane);
      if (n < 200)
        for (int v = 0; v < 8; ++v) {
          int m = m0 + v + 8 * half;
          float a = tanhf(acc[v]);
          float z = s_z[m * 200 + n];
          float h = s_h[m * 200 + n];
          s_h[m * 200 + n] = (1.f - z) * h + z * a;
        }
    }
    __syncthreads();

    // Phase 4: t1 = elu(h @ Wt1 + bt1) -> s_x
    for (int tile = wave; tile < 52; tile += 32) {
      int m0 = (tile & 3) << 4, nt = tile >> 2;
      int n = (nt << 4) + lm;
      float bias = (n < 200) ? bt1[n] : 0.f;
      v8f acc; for (int v = 0; v < 8; ++v) acc[v] = bias;
      acc = frag_mm(s_h, 200, m0, wt1_p, 50, nt, acc, lm, half, lane);
      if (n < 200)
        for (int v = 0; v < 8; ++v)
          s_x[(m0 + v + 8 * half) * 200 + n] = elu1(acc[v]);
    }
    __syncthreads();

    // Phase 5+6 (independent): y = t1@Wt2+bt2 -> s_y
    //                          r1 = elu(h@Wr1h + eproj + br1) -> s_rh
    for (int tile = wave; tile < 68; tile += 32) {
      if (tile < 16) {
        int m0 = (tile & 3) << 4, nt = tile >> 2;
        int n = (nt << 4) + lm;
        float bias = (n < 60) ? bt2[n] : 0.f;
        v8f acc; for (int v = 0; v < 8; ++v) acc[v] = bias;
        acc = frag_mm(s_x, 200, m0, wt2_p, 50, nt, acc, lm, half, lane);
        for (int v = 0; v < 8; ++v)
          s_y[(m0 + v + 8 * half) * 64 + n] = acc[v];
      } else {
        int t2 = tile - 16;
        int m0 = (t2 & 3) << 4, nt = t2 >> 2;
        int n = (nt << 4) + lm;
        v8f acc;
        for (int v = 0; v < 8; ++v) {
          int m = m0 + v + 8 * half;
          acc[v] = (n < 200)
                     ? eproj[((size_t)(m * T_ + t)) * HID_ + n] + br1[n]
                     : 0.f;
        }
        acc = frag_mm(s_h, 200, m0, wr1h_p, 50, nt, acc, lm, half, lane);
        if (n < 200)
          for (int v = 0; v < 8; ++v)
            s_rh[(m0 + v + 8 * half) * 200 + n] = elu1(acc[v]);
      }
    }
    __syncthreads();

    // Phase 7: yq = r1 @ Wr2 + br2 -> s_yq
    for (int tile = wave; tile < 16; tile += 32) {
      int m0 = (tile & 3) << 4, nt = tile >> 2;
      int n = (nt << 4) + lm;
      float bias = (n < 60) ? br2[n] : 0.f;
      v8f acc; for (int v = 0; v < 8; ++v) acc[v] = bias;
      acc = frag_mm(s_rh, 200, m0, wr2_p, 50, nt, acc, lm, half, lane);
      for (int v = 0; v < 8; ++v)
        s_yq[(m0 + v + 8 * half) * 64 + n] = acc[v];
    }
    __syncthreads();

    // Phase 8: epilogue — next stoch, and pack output row
    for (int idx = tid; idx < B_ * STOCH_; idx += 1024) {
      int b = idx / STOCH_, j = idx % STOCH_;
      float mq = s_yq[b * 64 + j];
      float sq = sp1(s_yq[b * 64 + STOCH_ + j]) + 0.1f;
      s_stoch[b * 48 + j] = mq + sq * npost[((size_t)(b * T_ + t)) * STOCH_ + j];
    }
    for (int idx = tid; idx < B_ * OUTC_; idx += 1024) {
      int b = idx / OUTC_, c = idx % OUTC_;
      float val;
      if (c < 30) {
        val = s_yq[b * 64 + c];                                   // mean_q
      } else if (c < 60) {
        val = sp1(s_yq[b * 64 + c]) + 0.1f;                       // std_q
      } else if (c < 90) {
        int j = c - 60;
        val = s_yq[b * 64 + j] +
              (sp1(s_yq[b * 64 + 30 + j]) + 0.1f) *
                  npost[((size_t)(b * T_ + t)) * STOCH_ + j];     // stoch_q
      } else if (c < 290) {
        val = s_h[b * 200 + (c - 90)];                            // deter
      } else if (c < 320) {
        val = s_y[b * 64 + (c - 290)];                            // mean_p
      } else if (c < 350) {
        val = sp1(s_y[b * 64 + (c - 290)]) + 0.1f;                // std_p
      } else if (c < 380) {
        int j = c - 350;
        val = s_y[b * 64 + j] +
              (sp1(s_y[b * 64 + 30 + j]) + 0.1f) *
                  nprior[((size_t)(b * T_ + t)) * STOCH_ + j];    // stoch_p
      } else {
        val = s_h[b * 200 + (c - 380)];                           // deter
      }
      out[((size_t)(b * T_ + t)) * OUTC_ + c] = val;
    }
    __syncthreads();
  }
}

// ---------------------------------------------------------------------------
extern "C" void kernel_launch(void* const* d_in, const int* in_sizes, int n_in,
                              void* d_out, int out_size, void* d_ws, size_t ws_size,
                              hipStream_t stream) {
  const float* embed  = (const float*)d_in[0];
  const float* action = (const float*)d_in[1];
  const float* nprior = (const float*)d_in[2];
  const float* npost  = (const float*)d_in[3];
  const float* We  = (const float*)d_in[4];
  const float* be  = (const float*)d_in[5];
  const float* wi  = (const float*)d_in[6];
  const float* wh  = (const float*)d_in[7];
  const float* bg  = (const float*)d_in[8];
  const float* Wt1 = (const float*)d_in[9];
  const float* bt1 = (const float*)d_in[10];
  const float* Wt2 = (const float*)d_in[11];
  const float* bt2 = (const float*)d_in[12];
  const float* Wr1 = (const float*)d_in[13];
  const float* br1 = (const float*)d_in[14];
  const float* Wr2 = (const float*)d_in[15];
  const float* br2 = (const float*)d_in[16];

  float* ws     = (float*)d_ws;
  float* eproj  = ws + EPROJ_OFF;
  float* we_p   = ws + WE_P_OFF;
  float* wi_p   = ws + WI_P_OFF;
  float* wh_p   = ws + WH_P_OFF;
  float* wt1_p  = ws + WT1_P_OFF;
  float* wt2_p  = ws + WT2_P_OFF;
  float* wr1h_p = ws + WR1H_P_OFF;
  float* wr1e_p = ws + WR1E_P_OFF;
  float* wr2_p  = ws + WR2_P_OFF;

  auto rp = [&](const float* W, float* Wp, int K, int rowOff, int ldw, int N,
                int Kg, int Nt) {
    int total = Nt * Kg * 64;
    int blocks = (total + 255) / 256;
    repack_w<<<blocks, 256, 0, stream>>>(W, Wp, K, rowOff, ldw, N, Kg, Nt);
  };
  rp(We,  we_p,   42,   0,  HID_, HID_, 11, 13);   // [stoch|act] -> hid
  rp(wi,  wi_p,   HID_, 0,  600,  600,  50, 38);
  rp(wh,  wh_p,   DETER_, 0, 600, 600,  50, 38);
  rp(Wt1, wt1_p,  DETER_, 0, HID_, HID_, 50, 13);
  rp(Wt2, wt2_p,  HID_, 0,  60,   60,   50, 4);
  rp(Wr1, wr1h_p, DETER_, 0, HID_, HID_, 50, 13);  // deter half
  rp(Wr1, wr1e_p, EDIM_, DETER_, HID_, HID_, 256, 13); // embed half
  rp(Wr2, wr2_p,  HID_, 0,  60,   60,   50, 4);

  // embed projection: 1024x13 tiles, 8 waves per 256-thread block
  eproj_kernel<<<(1024 * 13) / 8, 256, 0, stream>>>(embed, wr1e_p, eproj);

  // persistent sequential scan: 1 workgroup, 32 waves, 250KB dynamic LDS
  rssm_seq<<<1, 1024, SMEM_FLOATS * sizeof(float), stream>>>(
      action, nprior, npost, be, bg, bt1, bt2, br1, br2,
      we_p, wi_p, wh_p, wt1_p, wt2_p, wr1h_p, wr2_p, eproj, (float*)d_out);
}